// SAGEConvManual_352187319164
// MI455X (gfx1250) — compile-verified
//
#include <hip/hip_runtime.h>
#include <math.h>

#define N_NODES 50000
#define N_EDGES 800000
#define IN_CH   128
#define OUT_CH  128
#define K_TOT   256            // 2*IN_CH
#define LDS_STRIDE 260         // 256 + 4 pad -> conflict-free b64 A-operand reads
#define OUT_STRIDE 132         // 128 + 4 pad

typedef __attribute__((ext_vector_type(2))) float v2f;
typedef __attribute__((ext_vector_type(8))) float v8f;

// ---------------------------------------------------------------- zero scratch
__global__ void sage_zero_kernel(float* __restrict__ p, long n) {
    long i = (long)blockIdx.x * blockDim.x + threadIdx.x;
    if (i < n) p[i] = 0.0f;
}

// ------------------------------------------------- W[128,256] -> Wt[256,128]
__global__ void sage_transpose_w_kernel(const float* __restrict__ W,
                                        float* __restrict__ Wt) {
    int i = blockIdx.x * blockDim.x + threadIdx.x;
    if (i < K_TOT * OUT_CH) {
        int k = i / OUT_CH;      // 0..255
        int n = i % OUT_CH;      // 0..127
        Wt[i] = W[n * K_TOT + k];
    }
}

// ------------------------------------------------ scatter-add: one wave/edge
__global__ __launch_bounds__(256)
void sage_scatter_kernel(const float* __restrict__ x,
                         const long long* __restrict__ ei,
                         float* __restrict__ neigh,
                         float* __restrict__ deg) {
    int gtid = blockIdx.x * blockDim.x + threadIdx.x;
    int edge = gtid >> 5;
    int lane = gtid & 31;
    if (edge >= N_EDGES) return;
    long long src = ei[edge];
    long long dst = ei[N_EDGES + edge];
    const float* xs = x + (long)src * IN_CH;
    float* nd = neigh + (long)dst * IN_CH;
#pragma unroll
    for (int i = 0; i < 4; ++i) {
        int c = lane + 32 * i;                 // contiguous 128B per instruction
        atomicAdd(nd + c, xs[c]);
    }
    if (lane == 0) atomicAdd(deg + dst, 1.0f);
}

// --------------------------- fused GEMM (WMMA f32) + bias + row L2-normalize
__global__ __launch_bounds__(256)
void sage_gemm_kernel(const float* __restrict__ x,
                      const float* __restrict__ neigh,
                      const float* __restrict__ deg,
                      const float* __restrict__ Wt,   // [K_TOT][OUT_CH]
                      const float* __restrict__ bias,
                      float* __restrict__ out) {
    __shared__ float hs[16 * LDS_STRIDE];     // H tile: 16 rows x 256 (padded)
    __shared__ float outs[16 * OUT_STRIDE];   // 16 x 128 result (padded)
    __shared__ float rnorm[16];

    const int row0 = blockIdx.x * 16;         // 50000/16 = 3125 exact
    const int tid  = threadIdx.x;
    const int wave = tid >> 5;                // 8 waves, one 16-col tile each
    const int lane = tid & 31;
    const int half = lane >> 4;               // 0: K pair {0,1} | 1: K pair {2,3}
    const int l15  = lane & 15;

    // Stage H = [x_row | neigh_row/deg] into LDS; j-th pass loads full row j.
    for (int j = 0; j < 16; ++j) {
        int node = row0 + j;
        int c = tid;                           // 0..255, coalesced
        float v;
        if (c < IN_CH) {
            v = x[(long)node * IN_CH + c];
        } else {
            float d = deg[node];
            d = d > 1.0f ? d : 1.0f;
            v = neigh[(long)node * IN_CH + (c - IN_CH)] / d;
        }
        hs[j * LDS_STRIDE + c] = v;
    }
    __syncthreads();

    // 64 chained V_WMMA_F32_16X16X4_F32 over K = 256.
    // A layout: lane l holds A[M=l&15][K = k0 + 2*(l>>4) + {0,1}]
    // B layout: lane l holds B[K = k0 + 2*(l>>4) + {0,1}][N = l&15]
    const int n0 = wave * 16;
    const float* wtb = Wt + n0 + l15;
    v8f acc = {};
#pragma unroll 4
    for (int k0 = 0; k0 < K_TOT; k0 += 4) {
        int ka = k0 + half * 2;
        v2f a;
        a.x = hs[l15 * LDS_STRIDE + ka];
        a.y = hs[l15 * LDS_STRIDE + ka + 1];
        v2f bb;
        bb.x = wtb[(long)ka * OUT_CH];
        bb.y = wtb[(long)(ka + 1) * OUT_CH];
        acc = __builtin_amdgcn_wmma_f32_16x16x4_f32(
            /*neg_a=*/false, a, /*neg_b=*/false, bb,
            /*c_mod=*/(short)0, acc, /*reuse_a=*/false, /*reuse_b=*/false);
    }

    // C/D layout: lane l, VGPR r -> out[M = r + 8*(l>>4)][N = l&15]
#pragma unroll
    for (int r = 0; r < 8; ++r) {
        int m = r + half * 8;
        int n = n0 + l15;
        outs[m * OUT_STRIDE + n] = acc[r] + bias[n];
    }
    __syncthreads();

    // Row L2 norms (16 rows, one thread each; tiny).
    if (tid < 16) {
        float s = 0.0f;
#pragma unroll 8
        for (int c = 0; c < OUT_CH; ++c) {
            float v = outs[tid * OUT_STRIDE + c];
            s += v * v;
        }
        float nrm = sqrtf(s);
        rnorm[tid] = 1.0f / fmaxf(nrm, 1e-12f);
    }
    __syncthreads();

    // Write 16x128 normalized outputs, coalesced.
    for (int j = 0; j < 8; ++j) {
        int idx = tid + 256 * j;               // 2048 elems
        int m = idx >> 7;
        int c = idx & 127;
        out[(long)(row0 + m) * OUT_CH + c] = outs[m * OUT_STRIDE + c] * rnorm[m];
    }
}

extern "C" void kernel_launch(void* const* d_in, const int* in_sizes, int n_in,
                              void* d_out, int out_size, void* d_ws, size_t ws_size,
                              hipStream_t stream) {
    (void)in_sizes; (void)n_in; (void)out_size; (void)ws_size;
    const float*     x  = (const float*)d_in[0];
    const long long* ei = (const long long*)d_in[1];   // int64 edge_index
    const float*     W  = (const float*)d_in[2];
    const float*     b  = (const float*)d_in[3];
    float*           out = (float*)d_out;

    // Workspace layout: neigh [50000*128] | deg [50000] | Wt [256*128]
    float* neigh = (float*)d_ws;
    float* deg   = neigh + (size_t)N_NODES * IN_CH;
    float* Wt    = deg + N_NODES;

    long nz = (long)N_NODES * IN_CH + N_NODES;
    sage_zero_kernel<<<(nz + 255) / 256, 256, 0, stream>>>(neigh, nz);
    sage_transpose_w_kernel<<<(K_TOT * OUT_CH + 255) / 256, 256, 0, stream>>>(W, Wt);

    long sthreads = (long)N_EDGES * 32;                // one wave32 per edge
    sage_scatter_kernel<<<(sthreads + 255) / 256, 256, 0, stream>>>(x, ei, neigh, deg);

    sage_gemm_kernel<<<N_NODES / 16, 256, 0, stream>>>(x, neigh, deg, Wt, b, out);
}